// Attention_28913719836956
// MI455X (gfx1250) — compile-verified
//
#include <hip/hip_runtime.h>

typedef __attribute__((ext_vector_type(2))) float v2f;
typedef __attribute__((ext_vector_type(8))) float v8f;
typedef __attribute__((ext_vector_type(4))) unsigned int u32x4;
typedef __attribute__((ext_vector_type(8))) int i32x8;
typedef __attribute__((ext_vector_type(4))) int i32x4;

#define SEGLEN 256
#define DHEAD  64
#define NSEQ   4096
#define NSEGS  16
#define LSTR   66   // LDS row stride (floats): even (aligned v2f) + bank-skewed

__device__ __forceinline__ v8f wmma4(v2f a, v2f b, v8f c) {
  // D(16x16,f32) = A(16x4,f32) * B(4x16,f32) + C     -> v_wmma_f32_16x16x4_f32
  return __builtin_amdgcn_wmma_f32_16x16x4_f32(false, a, false, b, (short)0, c,
                                               false, false);
}

__device__ __forceinline__ float squash(float x) {  // elu(x)+1
  return x > 0.f ? x + 1.f : __expf(x);
}

// Tensor Data Mover: 2D fp32 tile -> LDS, padding rows of 64 dwords by 2 dwords
// so rows land at stride LSTR=66 in LDS. (D# per CDNA5 ISA)
// This toolchain exposes the 6-arg builtin: (g0, g1, g2, g3, g_extra, cpol).
__device__ __forceinline__ void tdm_load_2d_pad66(unsigned lds_off, const void* gptr,
                                                  unsigned dim0, unsigned dim1,
                                                  unsigned stride0,
                                                  unsigned tile0, unsigned tile1) {
  unsigned long long ga = (unsigned long long)(uintptr_t)gptr;
  u32x4 g0;
  g0[0] = 1u;                                             // count=1, user desc
  g0[1] = lds_off;                                        // lds_addr (bytes)
  g0[2] = (unsigned)(ga & 0xffffffffull);                 // global_addr[31:0]
  g0[3] = (unsigned)((ga >> 32) & 0x01ffffffull)          // global_addr[56:32]
        | 0x80000000u;                                    // type=2 ("image")
  i32x8 g1;
  g1[0] = (int)((2u << 16)     // data_size = 4B
              | (1u << 20)     // pad_enable
              | (5u << 22)     // pad_interval: 2^(5+1)=64 dwords
              | (1u << 25));   // pad_amount: 2 dwords
  g1[1] = (int)((dim0 & 0xffffu) << 16);                              // tensor_dim0 lo
  g1[2] = (int)(((dim0 >> 16) & 0xffffu) | ((dim1 & 0xffffu) << 16)); // dim0 hi|dim1 lo
  g1[3] = (int)(((dim1 >> 16) & 0xffffu) | ((tile0 & 0xffffu) << 16));// dim1 hi|tile0
  g1[4] = (int)(tile1 & 0xffffu);                                     // tile1 (tile2=0)
  g1[5] = (int)stride0;                                               // dim0_stride lo
  g1[6] = 0;
  g1[7] = 0;
  i32x4 zg4 = {0, 0, 0, 0};
  i32x8 zg8 = {0, 0, 0, 0, 0, 0, 0, 0};
  __builtin_amdgcn_tensor_load_to_lds(g0, g1, zg4, zg4, zg8, 0);
}

// =====================================================================
// K1: per (b,h,seg): causal flash SDPA -> out = (1-g)*attn ; and segment
//     deltas dM = (elu(k)+1)^T @ v  [64x64], dn = sum_n (elu(k)+1) [64]
// =====================================================================
__global__ __launch_bounds__(256, 1)
void k1_attn_delta_28913719836956(const float* __restrict__ qg,
                                  const float* __restrict__ kg,
                                  const float* __restrict__ vg,
                                  const float* __restrict__ gate,
                                  float* __restrict__ out,
                                  float* __restrict__ wsM,
                                  float* __restrict__ wsN) {
  __shared__ float Qs[SEGLEN][LSTR];
  __shared__ float Ks[SEGLEN][LSTR];
  __shared__ float Vs[SEGLEN][LSTR];
  __shared__ float SKs[SEGLEN][LSTR];
  __shared__ float invf[32];
  __shared__ float Pt[8][16][18];

  const int tid  = threadIdx.x;
  const int lane = tid & 31;
  const int wv   = tid >> 5;
  const int half = lane >> 4;
  const int l16  = lane & 15;
  const int bh   = blockIdx.x >> 4;
  const int seg  = blockIdx.x & 15;
  const int h    = bh & 7;
  const v8f z = {0.f,0.f,0.f,0.f,0.f,0.f,0.f,0.f};

  const size_t headbase = (size_t)bh * NSEQ * DHEAD;

  if (tid < 32) invf[tid] = __expf(-(float)tid * 0.28782313662425572f); // ln(1e4)/32
  if (wv == 0) {  // async DMA: V segment -> Vs while we do RoPE math
    tdm_load_2d_pad66((unsigned)(uintptr_t)&Vs[0][0],
                      vg + headbase + (size_t)seg * SEGLEN * DHEAD,
                      DHEAD, NSEQ, DHEAD, DHEAD, SEGLEN);
  }
  __syncthreads();   // invf visible

  const float g1m = 1.f - 1.f / (1.f + __expf(-gate[h]));  // (1 - sigmoid(gate))

  // ---- RoPE: one row per thread ----
  {
    const int n = tid, pos = seg * SEGLEN + n;
    float cs[32], sn[32];
    #pragma unroll
    for (int j = 0; j < 32; ++j) __sincosf((float)pos * invf[j], &sn[j], &cs[j]);
    float x[64];
    const float* qrow = qg + headbase + (size_t)pos * DHEAD;
    #pragma unroll
    for (int i = 0; i < 16; ++i) {
      float4 a4 = reinterpret_cast<const float4*>(qrow)[i];
      x[4*i+0]=a4.x; x[4*i+1]=a4.y; x[4*i+2]=a4.z; x[4*i+3]=a4.w;
    }
    #pragma unroll
    for (int j = 0; j < 32; ++j) {
      Qs[n][j]    = x[j]*cs[j]    - x[j+32]*sn[j];
      Qs[n][j+32] = x[j+32]*cs[j] + x[j]*sn[j];
    }
    const float* krow = kg + headbase + (size_t)pos * DHEAD;
    #pragma unroll
    for (int i = 0; i < 16; ++i) {
      float4 a4 = reinterpret_cast<const float4*>(krow)[i];
      x[4*i+0]=a4.x; x[4*i+1]=a4.y; x[4*i+2]=a4.z; x[4*i+3]=a4.w;
    }
    #pragma unroll
    for (int j = 0; j < 32; ++j) {
      float r0 = x[j]*cs[j]    - x[j+32]*sn[j];
      float r1 = x[j+32]*cs[j] + x[j]*sn[j];
      Ks[n][j] = r0;        Ks[n][j+32] = r1;
      SKs[n][j] = squash(r0); SKs[n][j+32] = squash(r1);
    }
  }
  if (wv == 0) __builtin_amdgcn_s_wait_tensorcnt(0);  // Vs DMA complete
  __syncthreads();

  // ---- flash attention: each wave owns 2 query row-blocks ----
  #pragma unroll
  for (int rbi = 0; rbi < 2; ++rbi) {
    const int rb   = wv * 2 + rbi;
    const int qrow = rb * 16 + l16;
    float m[8], lsum[8];
    v8f o[4];
    #pragma unroll
    for (int r = 0; r < 8; ++r) { m[r] = -1e30f; lsum[r] = 0.f; }
    #pragma unroll
    for (int ct = 0; ct < 4; ++ct) o[ct] = z;

    for (int j = 0; j <= rb; ++j) {
      v8f s = z;
      #pragma unroll
      for (int k0 = 0; k0 < DHEAD; k0 += 4) {        // S = Q K^T
        v2f a = *(const v2f*)&Qs[qrow][k0 + 2*half];
        v2f b = *(const v2f*)&Ks[j*16 + l16][k0 + 2*half];
        s = wmma4(a, b, s);
      }
      float sf[8];
      #pragma unroll
      for (int r = 0; r < 8; ++r) {
        float sv = s[r] * 0.125f;
        if (j == rb && l16 > r + 8*half) sv = -1e30f;
        float rm = sv;
        rm = fmaxf(rm, __shfl_xor(rm, 1));
        rm = fmaxf(rm, __shfl_xor(rm, 2));
        rm = fmaxf(rm, __shfl_xor(rm, 4));
        rm = fmaxf(rm, __shfl_xor(rm, 8));
        float mn  = fmaxf(m[r], rm);
        float scl = __expf(m[r] - mn);
        float p   = __expf(sv - mn);
        float rs  = p;
        rs += __shfl_xor(rs, 1);
        rs += __shfl_xor(rs, 2);
        rs += __shfl_xor(rs, 4);
        rs += __shfl_xor(rs, 8);
        lsum[r] = lsum[r] * scl + rs;
        m[r] = mn; sf[r] = scl;
        Pt[wv][r + 8*half][l16] = p;
      }
      #pragma unroll
      for (int ct = 0; ct < 4; ++ct) {               // O = O*scl + P @ V
        v8f acc = o[ct];
        #pragma unroll
        for (int r = 0; r < 8; ++r) acc[r] *= sf[r];
        #pragma unroll
        for (int k0 = 0; k0 < 16; k0 += 4) {
          v2f a = *(const v2f*)&Pt[wv][l16][k0 + 2*half];
          v2f b;
          b.x = Vs[j*16 + k0 + 2*half + 0][ct*16 + l16];
          b.y = Vs[j*16 + k0 + 2*half + 1][ct*16 + l16];
          acc = wmma4(a, b, acc);
        }
        o[ct] = acc;
      }
    }
    const size_t outbase = headbase + (size_t)(seg*SEGLEN + rb*16) * DHEAD;
    #pragma unroll
    for (int ct = 0; ct < 4; ++ct) {
      #pragma unroll
      for (int r = 0; r < 8; ++r) {
        const int row = r + 8*half;
        out[outbase + (size_t)row*DHEAD + ct*16 + l16] = g1m * (o[ct][r] / lsum[r]);
      }
    }
  }

  // ---- segment memory delta: dM = SK^T @ V (no barrier needed: reads only) ----
  float* dM = wsM + (size_t)(bh*NSEGS + seg) * DHEAD * DHEAD;
  #pragma unroll
  for (int ti = 0; ti < 2; ++ti) {
    const int t  = wv * 2 + ti;
    const int d0 = (t >> 2) * 16;
    const int e0 = (t & 3) * 16;
    v8f c = z;
    #pragma unroll 4
    for (int n0 = 0; n0 < SEGLEN; n0 += 4) {
      v2f a, b;
      a.x = SKs[n0 + 2*half + 0][d0 + l16];
      a.y = SKs[n0 + 2*half + 1][d0 + l16];
      b.x = Vs[n0 + 2*half + 0][e0 + l16];
      b.y = Vs[n0 + 2*half + 1][e0 + l16];
      c = wmma4(a, b, c);
    }
    #pragma unroll
    for (int r = 0; r < 8; ++r)
      dM[(size_t)(d0 + r + 8*half)*DHEAD + e0 + l16] = c[r];
  }
  if (tid < DHEAD) {
    float acc = 0.f;
    #pragma unroll 8
    for (int n = 0; n < SEGLEN; ++n) acc += SKs[n][tid];
    wsN[(size_t)(bh*NSEGS + seg)*DHEAD + tid] = acc;
  }
}

// =====================================================================
// K2: in-place exclusive prefix over segments of dM / dn (per b,h)
// =====================================================================
__global__ __launch_bounds__(256, 1)
void k2_prefix_28913719836956(float* __restrict__ wsM, float* __restrict__ wsN) {
  const int tid = threadIdx.x;
  const int bh  = blockIdx.x;
  float rm[16];
  #pragma unroll
  for (int j = 0; j < 16; ++j) rm[j] = 0.f;
  float rn = 0.f;
  for (int seg = 0; seg < NSEGS; ++seg) {
    float* p = wsM + (size_t)(bh*NSEGS + seg) * DHEAD * DHEAD;
    #pragma unroll
    for (int j = 0; j < 16; ++j) {
      const int idx = tid + 256*j;
      float t = p[idx];
      p[idx] = rm[j];
      rm[j] += t;
    }
    if (tid < DHEAD) {
      float* pn = wsN + (size_t)(bh*NSEGS + seg)*DHEAD + tid;
      float t = *pn;
      *pn = rn;
      rn += t;
    }
  }
}

// =====================================================================
// K3: per (b,h,seg): mem_out = (elu(q)+1) @ M_prefix / ((elu(q)+1)·n_prefix+1e-6)
//     out += g * mem_out
// =====================================================================
__global__ __launch_bounds__(256, 1)
void k3_memout_28913719836956(const float* __restrict__ qg,
                              const float* __restrict__ gate,
                              const float* __restrict__ wsM,
                              const float* __restrict__ wsN,
                              float* __restrict__ out) {
  __shared__ float SQs[SEGLEN][LSTR];
  __shared__ float Ms[DHEAD][LSTR];
  __shared__ float Norm[DHEAD];
  __shared__ float Den[SEGLEN];
  __shared__ float invf[32];

  const int tid  = threadIdx.x;
  const int lane = tid & 31;
  const int wv   = tid >> 5;
  const int half = lane >> 4;
  const int l16  = lane & 15;
  const int bh   = blockIdx.x >> 4;
  const int seg  = blockIdx.x & 15;
  const int h    = bh & 7;
  const v8f z = {0.f,0.f,0.f,0.f,0.f,0.f,0.f,0.f};

  const size_t headbase = (size_t)bh * NSEQ * DHEAD;

  if (tid < 32) invf[tid] = __expf(-(float)tid * 0.28782313662425572f);
  if (wv == 0) {  // async DMA: prefix memory [64x64] -> Ms (padded rows)
    tdm_load_2d_pad66((unsigned)(uintptr_t)&Ms[0][0],
                      wsM + (size_t)(bh*NSEGS + seg) * DHEAD * DHEAD,
                      DHEAD, DHEAD, DHEAD, DHEAD, DHEAD);
  }
  if (tid < DHEAD) Norm[tid] = wsN[(size_t)(bh*NSEGS + seg)*DHEAD + tid];
  __syncthreads();   // invf + Norm visible

  const float g = 1.f / (1.f + __expf(-gate[h]));

  // ---- RoPE + squash + denom: one row per thread ----
  {
    const int n = tid, pos = seg * SEGLEN + n;
    float cs[32], sn[32];
    #pragma unroll
    for (int j = 0; j < 32; ++j) __sincosf((float)pos * invf[j], &sn[j], &cs[j]);
    float x[64];
    const float* qrow = qg + headbase + (size_t)pos * DHEAD;
    #pragma unroll
    for (int i = 0; i < 16; ++i) {
      float4 a4 = reinterpret_cast<const float4*>(qrow)[i];
      x[4*i+0]=a4.x; x[4*i+1]=a4.y; x[4*i+2]=a4.z; x[4*i+3]=a4.w;
    }
    float acc = 1e-6f;
    #pragma unroll
    for (int j = 0; j < 32; ++j) {
      float r0 = squash(x[j]*cs[j]    - x[j+32]*sn[j]);
      float r1 = squash(x[j+32]*cs[j] + x[j]*sn[j]);
      SQs[n][j] = r0;   SQs[n][j+32] = r1;
      acc += r0 * Norm[j] + r1 * Norm[j+32];
    }
    Den[n] = acc;
  }
  if (wv == 0) __builtin_amdgcn_s_wait_tensorcnt(0);  // Ms DMA complete
  __syncthreads();

  // ---- mem_out via WMMA, blend into out ----
  #pragma unroll
  for (int rbi = 0; rbi < 2; ++rbi) {
    const int rb   = wv * 2 + rbi;
    const int qrow = rb * 16 + l16;
    v8f mo[4];
    #pragma unroll
    for (int ct = 0; ct < 4; ++ct) {
      v8f acc = z;
      #pragma unroll
      for (int k0 = 0; k0 < DHEAD; k0 += 4) {
        v2f a = *(const v2f*)&SQs[qrow][k0 + 2*half];
        v2f b;
        b.x = Ms[k0 + 2*half + 0][ct*16 + l16];
        b.y = Ms[k0 + 2*half + 1][ct*16 + l16];
        acc = wmma4(a, b, acc);
      }
      mo[ct] = acc;
    }
    const size_t outbase = headbase + (size_t)(seg*SEGLEN + rb*16) * DHEAD;
    #pragma unroll
    for (int ct = 0; ct < 4; ++ct) {
      #pragma unroll
      for (int r = 0; r < 8; ++r) {
        const int row = r + 8*half;
        const size_t oi = outbase + (size_t)row*DHEAD + ct*16 + l16;
        out[oi] = out[oi] + g * (mo[ct][r] / Den[rb*16 + row]);
      }
    }
  }
}

extern "C" void kernel_launch(void* const* d_in, const int* in_sizes, int n_in,
                              void* d_out, int out_size, void* d_ws, size_t ws_size,
                              hipStream_t stream) {
  (void)in_sizes; (void)n_in; (void)out_size; (void)ws_size;
  const float* q    = (const float*)d_in[0];
  const float* k    = (const float*)d_in[1];
  const float* v    = (const float*)d_in[2];
  const float* gate = (const float*)d_in[3];
  float* out = (float*)d_out;
  float* wsM = (float*)d_ws;                                   // 16*16*64*64 f32 = 4MB
  float* wsN = wsM + (size_t)16 * NSEGS * DHEAD * DHEAD;       // 16*16*64 f32 = 64KB

  k1_attn_delta_28913719836956<<<16*NSEGS, 256, 0, stream>>>(q, k, v, gate, out, wsM, wsN);
  k2_prefix_28913719836956<<<16, 256, 0, stream>>>(wsM, wsN);
  k3_memout_28913719836956<<<16*NSEGS, 256, 0, stream>>>(q, gate, wsM, wsN, out);
}